// RawTimesLayer_39290360824568
// MI455X (gfx1250) — compile-verified
//
#include <hip/hip_runtime.h>
#include <cstdint>

// Problem constants from the reference (W=48, TIME_COL=23, B=128, S=4096)
#define S_LEN   4096
#define W_COLS  48
#define TCOL    23                  // W/2 - 1
#define EXT     (TCOL + S_LEN)      // 4119 extended cumsum length
#define OUT_L   (EXT - (TCOL - 1))  // 4097 output columns per row
#define FILL_V  999.99f
#define NT      256                 // threads per block (8 waves)
#define PER     17                  // ceil(4119 / 256); 256*17 = 4352

typedef float v2f __attribute__((ext_vector_type(2)));
typedef float v8f __attribute__((ext_vector_type(8)));

// One block per batch row. Each thread owns a contiguous run of PER elements,
// computes 10^x locally, and the 256 per-thread partial sums are scanned with
// a WMMA triangular-matmul block scan (wave 0 only):
//   Y = T(16x16) * U(upper-tri ones)      -> per-row inclusive scans
//   Z = Lstrict(16x16) * bcast(rowtotals) -> cross-row prefix offsets
// each as 4 chained V_WMMA_F32_16X16X4_F32 (K=16 split into 4xK=4).
__global__ __launch_bounds__(NT) void raw_times_kernel(
    const float* __restrict__ data,      // (B, S, W) f32
    const int*   __restrict__ batch_N,   // (B,) int
    const float* __restrict__ batch_T0,  // (B,) f32
    float*       __restrict__ out)       // (B, 4097) f32
{
    __shared__ float ts[NT];     // per-thread partial sums
    __shared__ float incl[NT];   // inclusive scan of ts
    __shared__ float rowtot[16]; // row totals of Y

    const int b   = blockIdx.x;
    const int tid = threadIdx.x;
    const float* row0 = data + (size_t)b * (size_t)(S_LEN * W_COLS);

    // ---- local 10^x and per-thread serial sum -----------------------------
    float e[PER];
    const int base = tid * PER;
    float sum = 0.f;
#pragma unroll
    for (int k = 0; k < PER; ++k) {
        const int idx = base + k;
        if (idx < EXT) {
            const float v = (idx < TCOL)
                ? row0[idx]                                        // data[b,0,idx]
                : row0[(size_t)(idx - TCOL) * W_COLS + TCOL];      // data[b,idx-23,23]
            e[k] = exp10f(v);
        } else {
            e[k] = 0.f;  // padding contributes nothing to the scan
        }
        sum += e[k];
    }
    ts[tid] = sum;
    __syncthreads();

    // ---- WMMA block scan of the 256 partial sums (wave 0 only) ------------
    // T[r][c] = ts[16r + c].  A 16x4 f32 layout: lanes 0-15 rows, VGPR0 = K0
    // (lanes 0-15) | K2 (lanes 16-31), VGPR1 = K1 | K3.  C/D 16x16: VGPR v =
    // row v (lanes 0-15) / row v+8 (lanes 16-31), col = lane%16.
    v8f accY = {};
    v8f accZ = {};
    const int h = (tid >> 4) & 1;   // half-wave select
    const int n = tid & 15;         // A-row index / B,D column index

    if (tid < 32) {                 // wave 0, EXEC all ones inside
#pragma unroll
        for (int k = 0; k < 4; ++k) {
            const int k0 = 4 * k + 2 * h;
            v2f a, bb;
            a.x  = ts[16 * n + k0];               // A[r][k0]   (r == n here)
            a.y  = ts[16 * n + k0 + 1];           // A[r][k0+1]
            bb.x = (k0     <= n) ? 1.f : 0.f;     // U[k0][n]
            bb.y = (k0 + 1 <= n) ? 1.f : 0.f;     // U[k0+1][n]
            accY = __builtin_amdgcn_wmma_f32_16x16x4_f32(
                       false, a, false, bb, (short)0, accY, false, false);
        }
        if (n == 15) {                            // col 15 holds row totals
#pragma unroll
            for (int v = 0; v < 8; ++v) rowtot[v + 8 * h] = accY[v];
        }
    }
    __syncthreads();

    if (tid < 32) {
#pragma unroll
        for (int k = 0; k < 4; ++k) {
            const int k0 = 4 * k + 2 * h;
            v2f a, bb;
            a.x  = (k0     < n) ? 1.f : 0.f;      // Lstrict[r][k0]
            a.y  = (k0 + 1 < n) ? 1.f : 0.f;      // Lstrict[r][k0+1]
            bb.x = rowtot[k0];                    // broadcast across columns
            bb.y = rowtot[k0 + 1];
            accZ = __builtin_amdgcn_wmma_f32_16x16x4_f32(
                       false, a, false, bb, (short)0, accZ, false, false);
        }
        // incl[16*M + N] = Y[M][N] + Z[M],  M = v + 8h, N = n
#pragma unroll
        for (int v = 0; v < 8; ++v) {
            incl[16 * (v + 8 * h) + n] = accY[v] + accZ[v];
        }
    }
    __syncthreads();

    // ---- replay local run with exclusive prefix, write masked outputs -----
    const float prefix = (tid == 0) ? 0.f : incl[tid - 1];
    float run = prefix + batch_T0[b];
    const int Nb = batch_N[b];
    float* orow = out + (size_t)b * OUT_L;
#pragma unroll
    for (int k = 0; k < PER; ++k) {
        const int idx = base + k;
        run += e[k];
        if (idx >= (TCOL - 1) && idx < EXT) {
            const int j = idx - (TCOL - 1);       // output column 0..4096
            orow[j] = (j <= Nb) ? run : FILL_V;   // mask: j < N+1
        }
    }
}

extern "C" void kernel_launch(void* const* d_in, const int* in_sizes, int n_in,
                              void* d_out, int out_size, void* d_ws, size_t ws_size,
                              hipStream_t stream) {
    const float* data     = (const float*)d_in[0];
    const int*   batch_N  = (const int*)  d_in[1];
    const float* batch_T0 = (const float*)d_in[2];
    float*       out      = (float*)d_out;

    const int Bn = in_sizes[1];   // number of batch rows (128)
    raw_times_kernel<<<dim3(Bn), dim3(NT), 0, stream>>>(data, batch_N, batch_T0, out);
}